// HierarchicalSampling_13967233646657
// MI455X (gfx1250) — compile-verified
//
#include <hip/hip_runtime.h>
#include <stdint.h>

#define EPS 1e-5f

typedef float v2f __attribute__((ext_vector_type(2)));
typedef float v8f __attribute__((ext_vector_type(8)));

constexpr int NB = 64;        // bin edges per ray
constexpr int NW = 63;        // weights per ray
constexpr int NS = 128;       // samples per ray
constexpr int WSTRIDE = 66;   // LDS row stride for CDF (bank-conflict-free column reads)
constexpr int BSTRIDE = 64;   // LDS row stride for bins (16B-aligned rows for async b128)
constexpr int WAVES = 4;      // wave32s per block
constexpr int RAYS_PER_WAVE = 16;
constexpr int RAYS_PER_BLOCK = WAVES * RAYS_PER_WAVE;   // 64

// ---- CDNA5 async global->LDS path (ASYNCcnt-tracked DMA), with fallback ----
#if defined(__has_builtin)
#if __has_builtin(__builtin_amdgcn_global_load_async_to_lds_b32) && \
    __has_builtin(__builtin_amdgcn_global_load_async_to_lds_b128)
#define HAVE_ASYNC_LDS 1
#endif
#endif

// Exact parameter types per hipcc diagnostics:
//   b32 : (int addrspace(1)*, <lds ptr>, Ii offset, Ii cpol)
//   b128: (vector_size(16) int addrspace(1)*, <lds ptr>, Ii offset, Ii cpol)
typedef int v4i __attribute__((vector_size(16)));
typedef __attribute__((address_space(1))) int  g_i32;
typedef __attribute__((address_space(3))) int  l_i32;
typedef __attribute__((address_space(1))) v4i  g_i128;
typedef __attribute__((address_space(3))) v4i  l_i128;

static __device__ __forceinline__ void async_lds_b32(const void* g, void* l) {
#ifdef HAVE_ASYNC_LDS
    // flat->LDS addrspacecast keeps the low 32 bits on AMDGPU
    __builtin_amdgcn_global_load_async_to_lds_b32(
        (g_i32*)(uintptr_t)g, (l_i32*)(uint32_t)(uintptr_t)l, 0, 0);
#endif
}
static __device__ __forceinline__ void async_lds_b128(const void* g, void* l) {
#ifdef HAVE_ASYNC_LDS
    __builtin_amdgcn_global_load_async_to_lds_b128(
        (g_i128*)(uintptr_t)g, (l_i128*)(uint32_t)(uintptr_t)l, 0, 0);
#endif
}
static __device__ __forceinline__ void wait_asynccnt0() {
#if defined(__has_builtin)
#if __has_builtin(__builtin_amdgcn_s_wait_asynccnt)
    __builtin_amdgcn_s_wait_asynccnt(0);
    return;
#endif
#endif
    asm volatile("s_wait_asynccnt 0x0" ::: "memory");
}

// One wave32 owns 16 rays. Phases:
//  1) DMA raw weights (b32) + bins (b128) into LDS via GLOBAL_LOAD_ASYNC_TO_LDS,
//     prefetch u rows (global_prefetch_b8); s_wait_asynccnt; zero pad column 63
//  2) cumsum via V_WMMA_F32_16X16X4_F32: S(16x64) = W(16x64) x T(64x64),
//     T[k][j] = (k<=j && k<63) generated in-register; triangular chunk skip
//  3) scatter D tiles back over the weights region, adding the analytic EPS term
//     (cumsum(w+EPS) = cumsum(w) + (min(col,62)+1)*EPS); S[r][63] = total
//  4) 128 samples/ray: branchless 6-step binary search on raw CDF vs u*total in
//     LDS, interpolate with bin edges; coalesced u loads / out stores
__global__ __launch_bounds__(WAVES * 32)
void hs_sample_kernel(const float* __restrict__ bins,
                      const float* __restrict__ weights,
                      const float* __restrict__ u,
                      float* __restrict__ out)
{
    __shared__ float lds[WAVES * (RAYS_PER_WAVE * WSTRIDE + RAYS_PER_WAVE * BSTRIDE)];

    const int lane = threadIdx.x & 31;
    const int wave = threadIdx.x >> 5;
    float* Wl = lds + wave * (RAYS_PER_WAVE * WSTRIDE + RAYS_PER_WAVE * BSTRIDE);
    float* Bl = Wl + RAYS_PER_WAVE * WSTRIDE;
    const int rbase = blockIdx.x * RAYS_PER_BLOCK + wave * RAYS_PER_WAVE;

    // ---- phase 1: stage raw weights + bins into LDS; prefetch u ----
#ifdef HAVE_ASYNC_LDS
    for (int r = 0; r < RAYS_PER_WAVE; ++r) {
        const int ray = rbase + r;
        const float* wrow = weights + (long)ray * NW;
        async_lds_b32(wrow + lane, Wl + r * WSTRIDE + lane);
        if (lane < 31) {  // k2 = 32+lane in [32,62]: stays in-bounds of the 63 weights
            async_lds_b32(wrow + 32 + lane, Wl + r * WSTRIDE + 32 + lane);
        }
        __builtin_prefetch(u + (long)ray * NS + lane * 4, 0, 3);  // global_prefetch_b8
    }
    for (int rr = 0; rr < 8; ++rr) {          // bins: 2 rows per b128 op (16B/lane)
        const int row = rr * 2 + (lane >> 4);
        const int ray = rbase + row;
        async_lds_b128(bins + (long)ray * NB + (lane & 15) * 4,
                       Bl + row * BSTRIDE + (lane & 15) * 4);
    }
    wait_asynccnt0();
#else
    for (int r = 0; r < RAYS_PER_WAVE; ++r) {
        const int ray = rbase + r;
        const int k2 = lane + 32;
        Wl[r * WSTRIDE + lane] = weights[(long)ray * NW + lane];
        if (k2 < NW) Wl[r * WSTRIDE + k2] = weights[(long)ray * NW + k2];
        Bl[r * BSTRIDE + lane]      = bins[(long)ray * NB + lane];
        Bl[r * BSTRIDE + lane + 32] = bins[(long)ray * NB + lane + 32];
        __builtin_prefetch(u + (long)ray * NS + lane * 4, 0, 3);
    }
#endif
    if (lane < RAYS_PER_WAVE) Wl[lane * WSTRIDE + 63] = 0.0f;   // K=63 pad column
    __syncthreads();

    // ---- phase 2: inclusive prefix-sum as triangular matmul on the matrix pipe ----
    const int mrow  = lane & 15;    // M for A, N for B, column for C/D
    const int khalf = lane >> 4;    // 0: K+{0,1}, 1: K+{2,3}
    v8f acc[4] = {v8f{}, v8f{}, v8f{}, v8f{}};

#pragma unroll
    for (int c = 0; c < 16; ++c) {              // K chunks of 4
        const int k = c * 4 + khalf * 2;
        v2f a;                                   // A 16x4 f32 fragment
        a.x = Wl[mrow * WSTRIDE + k];
        a.y = Wl[mrow * WSTRIDE + k + 1];
#pragma unroll
        for (int t = 0; t < 4; ++t) {            // N tiles of 16 output bins
            if (t < (c >> 2)) continue;          // triangular: chunk needed only if k0 <= 16t+15
            const int n = t * 16 + mrow;
            v2f b;                               // B 4x16 fragment: T[k][n] = (k<=n && k<63)
            b.x = (k     <= n && k     < NW) ? 1.0f : 0.0f;
            b.y = (k + 1 <= n && k + 1 < NW) ? 1.0f : 0.0f;
            acc[t] = __builtin_amdgcn_wmma_f32_16x16x4_f32(
                false, a, false, b, (short)0, acc[t], false, false);
        }
    }

    // ---- phase 3: D tiles + analytic EPS -> LDS (raw CDF S over weights region) ----
#pragma unroll
    for (int t = 0; t < 4; ++t) {
        const int col = t * 16 + mrow;
        const float epsadd = (float)((col < 63 ? col : 62) + 1) * EPS;
#pragma unroll
        for (int v = 0; v < 8; ++v) {
            const int m = v + khalf * 8;                       // C/D row layout
            Wl[m * WSTRIDE + col] = acc[t][v] + epsadd;
        }
    }
    __syncthreads();

    // ---- phase 4: inverse-CDF sampling out of LDS ----
    for (int r = 0; r < RAYS_PER_WAVE; ++r) {
        const float* __restrict__ Srow = Wl + r * WSTRIDE;     // S[0..63], S[63]=total
        const float* __restrict__ Brow = Bl + r * BSTRIDE;
        const float total = Srow[63];
        const float invT  = 1.0f / total;
        const long  base  = (long)(rbase + r) * NS;
#pragma unroll
        for (int ss = 0; ss < 4; ++ss) {
            const int   s  = ss * 32 + lane;
            const float uu = u[base + s];
            const float ut = uu * total;
            // cdf[i] = (i==0) ? 0 : S[i-1]/total; find last i with cdf[i] <= uu
            // (equiv: i==0 || S[i-1] <= ut). j = lo+step never exceeds 63.
            int lo = 0;
#pragma unroll
            for (int step = 32; step >= 1; step >>= 1) {
                const int j = lo + step;
                lo = (Srow[j - 1] <= ut) ? j : lo;
            }
            const int below = lo;
            const int above = (lo + 1 < 64) ? (lo + 1) : 63;
            const float Slo = (below == 0) ? 0.0f : Srow[below - 1];
            const float Shi = Srow[above - 1];                 // above >= 1 always
            const float cdf_lo = Slo * invT;
            const float cdf_hi = Shi * invT;
            const float blo = Brow[below];
            const float bhi = Brow[above];
            float denom = cdf_hi - cdf_lo;
            denom = (denom < EPS) ? 1.0f : denom;
            const float tt = (uu - cdf_lo) / denom;
            out[base + s] = blo + tt * (bhi - blo);
        }
    }
}

extern "C" void kernel_launch(void* const* d_in, const int* in_sizes, int n_in,
                              void* d_out, int out_size, void* d_ws, size_t ws_size,
                              hipStream_t stream) {
    const float* bins    = (const float*)d_in[0];   // [R, 64]
    const float* weights = (const float*)d_in[1];   // [R, 63]
    const float* u       = (const float*)d_in[2];   // [R, 128]
    float* out = (float*)d_out;                     // [R, 128]

    const int R = in_sizes[0] / NB;                 // 262144
    const int blocks = R / RAYS_PER_BLOCK;          // 4096 (R divisible by 64)
    hs_sample_kernel<<<blocks, WAVES * 32, 0, stream>>>(bins, weights, u, out);
}